// ExpectedCaibrationError_35880156791393
// MI455X (gfx1250) — compile-verified
//
#include <hip/hip_runtime.h>
#include <hip/hip_bf16.h>
#include <stdint.h>

#define N_BINS 15

// ---------------------------------------------------------------------------
// CDNA5 (gfx1250) async global->LDS copy path.
//   global_load_async_to_lds_b128: per-lane, copies 16B from global addr
//   (VGPR pair) to LDS byte address (VGPR). Tracked with ASYNCcnt; per-wave
//   completion is IN ORDER, so s_wait_asynccnt <= K fences the oldest copies.
// ---------------------------------------------------------------------------
__device__ __forceinline__ void async_load_lds_b128(uint32_t lds_byte_addr,
                                                    const void* gaddr) {
  asm volatile("global_load_async_to_lds_b128 %0, %1, off"
               :: "v"(lds_byte_addr), "v"(gaddr)
               : "memory");
}
#define WAIT_ASYNCCNT(n) asm volatile("s_wait_asynccnt " #n ::: "memory")

__device__ __forceinline__ uint32_t lds_addr32(const void* p) {
  // Generic pointers into LDS: addr[31:0] is the LDS byte address (ISA 10.2).
  return (uint32_t)(uintptr_t)p;
}

// ---------------------------------------------------------------------------
// Shared helpers
// ---------------------------------------------------------------------------
__device__ __forceinline__ void bin_accumulate(float* s_accum, float conf,
                                               float acc) {
  // reference: lower < conf <= upper  =>  bin = ceil(conf*15) - 1
  int b = (int)ceilf(conf * (float)N_BINS) - 1;
  if (b >= 0) {
    if (b >= N_BINS) b = N_BINS - 1;
    atomicAdd(&s_accum[b], 1.0f);               // count
    atomicAdd(&s_accum[N_BINS + b], conf);      // sum conf
    atomicAdd(&s_accum[2 * N_BINS + b], acc);   // sum acc
  }
}

__device__ __forceinline__ void wave_argmax(float& v, int& idx) {
  #pragma unroll
  for (int off = 16; off > 0; off >>= 1) {
    float ov = __shfl_xor(v, off, 32);
    int   oi = __shfl_xor(idx, off, 32);
    // keep larger value; on tie keep smaller index (first occurrence)
    if (ov > v || (ov == v && oi < idx)) { v = ov; idx = oi; }
  }
}

// ---------------------------------------------------------------------------
// Main kernel, specialized for C == 100 (row = 400B), async-LDS double buffer
// ---------------------------------------------------------------------------
#define ROW_F 100
#define ROW_B (ROW_F * 4)                    // 400
#define TILE_ROWS 64
#define TILE_B (TILE_ROWS * ROW_B)           // 25600
#define MAIN_THREADS 320                     // 10 waves (wave32)
#define MAIN_WAVES (MAIN_THREADS / 32)
#define CHUNKS ((TILE_B / 16) / MAIN_THREADS)  // 5 x b128 per thread per tile
static_assert(CHUNKS * MAIN_THREADS * 16 == TILE_B, "tile must tile evenly");
static_assert(CHUNKS == 5, "s_wait_asynccnt immediate below assumes 5");

__launch_bounds__(MAIN_THREADS)
__global__ void ece_main_c100(const float* __restrict__ logits,
                              const int* __restrict__ labels,
                              int N, float* __restrict__ ws) {
  __shared__ float s_buf[2][TILE_B / 4];
  __shared__ float s_accum[3 * N_BINS];

  for (int i = threadIdx.x; i < 3 * N_BINS; i += MAIN_THREADS) s_accum[i] = 0.f;

  const int numTiles = (N + TILE_ROWS - 1) / TILE_ROWS;
  const size_t totalBytes = (size_t)N * ROW_B;
  const uint32_t ldsBase[2] = { lds_addr32(&s_buf[0][0]),
                                lds_addr32(&s_buf[1][0]) };
  const int lane = threadIdx.x & 31;
  const int wave = threadIdx.x >> 5;
  const char* gbase = (const char*)logits;

  auto issue_tile = [&](int tile, int buf) {
    const uint32_t lbase = ldsBase[buf];
    #pragma unroll
    for (int c = 0; c < CHUNKS; ++c) {
      const uint32_t elem = (uint32_t)(c * MAIN_THREADS + threadIdx.x) * 16u;
      size_t goff = (size_t)tile * TILE_B + elem;
      if (goff + 16 > totalBytes) goff = totalBytes - 16;  // uniform issue count
      async_load_lds_b128(lbase + elem, gbase + goff);
    }
  };

  int t = blockIdx.x;
  if (t < numTiles) issue_tile(t, 0);   // prologue prefetch into buffer 0
  __syncthreads();                      // also covers s_accum init

  int it = 0;
  for (; t < numTiles; t += (int)gridDim.x, ++it) {
    const int cur = it & 1;
    const int tn = t + (int)gridDim.x;
    if (tn < numTiles) {
      issue_tile(tn, cur ^ 1);          // prefetch next tile into other buffer
      WAIT_ASYNCCNT(5);                 // oldest tile's 5 copies are complete
    } else {
      WAIT_ASYNCCNT(0);
    }
    __syncthreads();                    // LDS tile visible to all waves

    const float* buf = s_buf[cur];
    const int rowBase = t * TILE_ROWS;
    for (int r = wave; r < TILE_ROWS; r += MAIN_WAVES) {
      const int row = rowBase + r;
      if (row >= N) break;              // uniform within wave
      const float* rp = buf + r * ROW_F;
      float v = -__builtin_inff();
      int idx = 0;
      for (int k = lane; k < ROW_F; k += 32) {
        const float x = rp[k];
        if (x > v) { v = x; idx = k; }  // keeps first occurrence within lane
      }
      wave_argmax(v, idx);
      if (lane == 0) {
        const float a = (idx == labels[row]) ? 1.0f : 0.0f;
        bin_accumulate(s_accum, v, a);
      }
    }
    __syncthreads();                    // buffer safe to overwrite next iter
  }

  for (int i = threadIdx.x; i < 3 * N_BINS; i += MAIN_THREADS)
    atomicAdd(&ws[i], s_accum[i]);
}

// ---------------------------------------------------------------------------
// Generic fallback for C != 100: wave-per-row, direct coalesced global loads
// ---------------------------------------------------------------------------
__global__ void ece_generic(const float* __restrict__ logits,
                            const int* __restrict__ labels,
                            int N, int C, float* __restrict__ ws) {
  __shared__ float s_accum[3 * N_BINS];
  for (int i = threadIdx.x; i < 3 * N_BINS; i += blockDim.x) s_accum[i] = 0.f;
  __syncthreads();

  const int lane = threadIdx.x & 31;
  const int wavesPerBlock = blockDim.x >> 5;
  const int gwave = blockIdx.x * wavesPerBlock + (threadIdx.x >> 5);
  const int nwave = gridDim.x * wavesPerBlock;

  for (int row = gwave; row < N; row += nwave) {
    const float* rp = logits + (size_t)row * C;
    float v = -__builtin_inff();
    int idx = 0;
    for (int k = lane; k < C; k += 32) {
      const float x = rp[k];
      if (x > v) { v = x; idx = k; }
    }
    wave_argmax(v, idx);
    if (lane == 0) {
      const float a = (idx == labels[row]) ? 1.0f : 0.0f;
      bin_accumulate(s_accum, v, a);
    }
  }
  __syncthreads();
  for (int i = threadIdx.x; i < 3 * N_BINS; i += blockDim.x)
    atomicAdd(&ws[i], s_accum[i]);
}

// ---------------------------------------------------------------------------
// Init + finalize
// ---------------------------------------------------------------------------
__global__ void ece_init(float* __restrict__ ws) {
  if (threadIdx.x < 3 * N_BINS) ws[threadIdx.x] = 0.f;
}

__global__ void ece_final(const float* __restrict__ ws,
                          float* __restrict__ out, int N) {
  if (threadIdx.x == 0 && blockIdx.x == 0) {
    const float n = (float)N;
    float ece = 0.f, acc = 0.f;
    for (int b = 0; b < N_BINS; ++b) {
      const float cnt = ws[b];
      if (cnt > 0.f) {
        const float prop = cnt / n;
        const float cb = ws[N_BINS + b] / cnt;
        const float ab = ws[2 * N_BINS + b] / cnt;
        ece += fabsf(cb - ab) * prop;
        acc += ab * prop;
      }
    }
    out[0] = ece * 100.f;
    out[1] = acc * 100.f;
  }
}

// ---------------------------------------------------------------------------
extern "C" void kernel_launch(void* const* d_in, const int* in_sizes, int n_in,
                              void* d_out, int out_size, void* d_ws,
                              size_t ws_size, hipStream_t stream) {
  const float* logits = (const float*)d_in[0];
  const int*   labels = (const int*)d_in[1];
  const int N = in_sizes[1];                      // 1,000,000
  const int C = in_sizes[1] ? in_sizes[0] / in_sizes[1] : 0;  // 100
  float* ws  = (float*)d_ws;                      // 45 floats of scratch
  float* out = (float*)d_out;                     // [ece, acc]

  ece_init<<<1, 64, 0, stream>>>(ws);

  if (C == ROW_F && N > 0) {
    const int numTiles = (N + TILE_ROWS - 1) / TILE_ROWS;
    const int blocks = numTiles < 4096 ? numTiles : 4096;
    ece_main_c100<<<blocks, MAIN_THREADS, 0, stream>>>(logits, labels, N, ws);
  } else if (N > 0) {
    ece_generic<<<1024, 256, 0, stream>>>(logits, labels, N, C, ws);
  }

  ece_final<<<1, 32, 0, stream>>>(ws, out, N);
}